// GraphConvolutionLayer_10307921510886
// MI455X (gfx1250) — compile-verified
//
#include <hip/hip_runtime.h>

typedef __attribute__((ext_vector_type(2))) float v2f;
typedef __attribute__((ext_vector_type(8))) float v8f;

#define D      256      // D_IN == D_OUT
#define BM     128      // block tile M
#define BN     64       // block tile N
#define BK     16       // K chunk
#define LDA    20       // padded LDS row stride for A tile (bank-conflict free, 16B-aligned rows)
#define LDBP   160      // LDS stride per k-pair for B tile (160 % 64banks == 32 -> halves disjoint)

// ---------------------------------------------------------------------------
// Kernel 1: support = x @ W  via V_WMMA_F32_16X16X4_F32
// 256 threads = 8 waves. Wave w computes rows [w*16, w*16+16) of the block's
// 128-row strip, across 4 n-tiles (64 cols) -> 4 x v8f accumulators.
// B tile stored k-pair interleaved: (k,n) -> Bs[(k>>1)*LDBP + n*2 + (k&1)],
// so each B fragment is a single aligned ds_load_b64 (no repack movs).
// ---------------------------------------------------------------------------
__global__ __launch_bounds__(256)
void gcn_gemm(const float* __restrict__ x, const float* __restrict__ w,
              float* __restrict__ support, int M) {
  __shared__ float As[BM * LDA];        // 128 x 16 fp32, padded stride 20
  __shared__ float Bs[(BK / 2) * LDBP]; // 8 k-pairs x 64 cols x 2, padded

  const int tid  = threadIdx.x;
  const int lane = tid & 31;
  const int wave = tid >> 5;       // 0..7 -> 16-row strip within block tile
  const int half = lane >> 4;      // 0/1: K-halves for A/B fragments
  const int l    = lane & 15;

  const int mBase = blockIdx.x * BM;
  const int nBase = blockIdx.y * BN;

  v8f acc0 = {}, acc1 = {}, acc2 = {}, acc3 = {};

  for (int kBase = 0; kBase < D; kBase += BK) {
    __syncthreads();
    // Stage A tile: 128x16 floats = 512 float4, 2 per thread. Zero-fill OOB
    // rows so WMMA can run with EXEC all-ones and stores alone are guarded.
#pragma unroll
    for (int i = 0; i < 2; ++i) {
      int f   = tid + i * 256;
      int row = f >> 2;
      int cf  = (f & 3) * 4;
      float4 v = make_float4(0.f, 0.f, 0.f, 0.f);
      int gm = mBase + row;
      if (gm < M)
        v = *(const float4*)(x + (size_t)gm * D + kBase + cf);
      *(float4*)(&As[row * LDA + cf]) = v;
    }
    // Stage B tile: 16x64 floats, 1 float4 per thread, scattered into the
    // k-pair-interleaved layout.
    {
      int row = tid >> 4;            // k within chunk: 0..15
      int cf  = (tid & 15) * 4;      // col: 0..60
      float4 v = *(const float4*)(w + (size_t)(kBase + row) * D + nBase + cf);
      float* bp = &Bs[(row >> 1) * LDBP + (row & 1)];
      bp[(cf + 0) * 2] = v.x;
      bp[(cf + 1) * 2] = v.y;
      bp[(cf + 2) * 2] = v.z;
      bp[(cf + 3) * 2] = v.w;
    }
    __syncthreads();

#pragma unroll
    for (int s = 0; s < 4; ++s) {
      const int k0 = s * 4 + 2 * half;  // always even; lanes 0-15: K=k0,k0+1 ; 16-31: +2
      v2f a;
      a.x = As[(wave * 16 + l) * LDA + k0];
      a.y = As[(wave * 16 + l) * LDA + k0 + 1];

      const float* bbase = &Bs[(k0 >> 1) * LDBP + l * 2];
      v2f b0 = *(const v2f*)(bbase +  0);   // n-tile 0: cols  0..15
      v2f b1 = *(const v2f*)(bbase + 32);   // n-tile 1: cols 16..31
      v2f b2 = *(const v2f*)(bbase + 64);   // n-tile 2: cols 32..47
      v2f b3 = *(const v2f*)(bbase + 96);   // n-tile 3: cols 48..63

      acc0 = __builtin_amdgcn_wmma_f32_16x16x4_f32(false, a, false, b0, (short)0, acc0, false, false);
      acc1 = __builtin_amdgcn_wmma_f32_16x16x4_f32(false, a, false, b1, (short)0, acc1, false, false);
      acc2 = __builtin_amdgcn_wmma_f32_16x16x4_f32(false, a, false, b2, (short)0, acc2, false, false);
      acc3 = __builtin_amdgcn_wmma_f32_16x16x4_f32(false, a, false, b3, (short)0, acc3, false, false);
    }
  }

  // C/D layout: VGPR r -> M = r (lanes 0-15) or r+8 (lanes 16-31), N = lane&15.
  const int col0 = nBase + l;
#pragma unroll
  for (int r = 0; r < 8; ++r) {
    int m = mBase + wave * 16 + r + 8 * half;
    if (m < M) {
      float* o = support + (size_t)m * D + col0;
      o[ 0] = acc0[r];
      o[16] = acc1[r];
      o[32] = acc2[r];
      o[48] = acc3[r];
    }
  }
}

// ---------------------------------------------------------------------------
// Kernel 2: out[i][:] = bias[:]   (one float4 per thread; d_out is poisoned)
// ---------------------------------------------------------------------------
__global__ __launch_bounds__(256)
void gcn_init_out(float4* __restrict__ out4, const float4* __restrict__ bias4,
                  int total4) {
  int i = blockIdx.x * blockDim.x + threadIdx.x;
  if (i < total4) out4[i] = bias4[i & 63];   // 256 floats = 64 float4 of bias
}

// ---------------------------------------------------------------------------
// Kernel 3: out[row[e]] += val[e] * support[col[e]]  — one wave32 per edge.
// Each lane handles 8 consecutive floats: 2x b128 gather (L2-resident; the
// 102 MB support array fits in the 192 MB L2) + 8 no-return
// global_atomic_add_f32 into the L2-resident out array.
// ---------------------------------------------------------------------------
__global__ __launch_bounds__(256)
void gcn_scatter(const float* __restrict__ support,
                 const float* __restrict__ edge_val,
                 const int* __restrict__ edge_row,
                 const int* __restrict__ edge_col,
                 float* __restrict__ out, int nEdges) {
  int wid = blockIdx.x * (blockDim.x >> 5) + (threadIdx.x >> 5);
  if (wid >= nEdges) return;
  int lane = threadIdx.x & 31;

  int   row = edge_row[wid];
  int   col = edge_col[wid];
  float val = edge_val[wid];

  const float4* src = (const float4*)(support + (size_t)col * D) + lane * 2;
  float4 a = src[0];
  float4 b = src[1];

  float* dst = out + (size_t)row * D + lane * 8;
  unsafeAtomicAdd(dst + 0, a.x * val);
  unsafeAtomicAdd(dst + 1, a.y * val);
  unsafeAtomicAdd(dst + 2, a.z * val);
  unsafeAtomicAdd(dst + 3, a.w * val);
  unsafeAtomicAdd(dst + 4, b.x * val);
  unsafeAtomicAdd(dst + 5, b.y * val);
  unsafeAtomicAdd(dst + 6, b.z * val);
  unsafeAtomicAdd(dst + 7, b.w * val);
}

// ---------------------------------------------------------------------------
extern "C" void kernel_launch(void* const* d_in, const int* in_sizes, int n_in,
                              void* d_out, int out_size, void* d_ws, size_t ws_size,
                              hipStream_t stream) {
  const float* x        = (const float*)d_in[0];   // [M, 256]
  const float* edge_val = (const float*)d_in[1];   // [E]
  const float* weight   = (const float*)d_in[2];   // [256, 256]
  const float* bias     = (const float*)d_in[3];   // [256]
  const int*   edge_row = (const int*)  d_in[4];   // [E]
  const int*   edge_col = (const int*)  d_in[5];   // [E]
  float*       out      = (float*)d_out;           // [M, 256]
  float*       support  = (float*)d_ws;            // [M, 256] scratch (102.4 MB)

  const int M = in_sizes[0] / D;   // 100000
  const int E = in_sizes[1];       // 3200000

  // 1) support = x @ W  (WMMA f32)
  dim3 gemmGrid((M + BM - 1) / BM, D / BN);
  gcn_gemm<<<gemmGrid, dim3(256), 0, stream>>>(x, weight, support, M);

  // 2) out = broadcast(bias)
  int total4 = M * (D / 4);
  gcn_init_out<<<(total4 + 255) / 256, dim3(256), 0, stream>>>(
      (float4*)out, (const float4*)bias, total4);

  // 3) scatter-add over edges (one wave per edge, 8 waves per block)
  int blocks = (E + 7) / 8;
  gcn_scatter<<<blocks, dim3(256), 0, stream>>>(support, edge_val, edge_row,
                                                edge_col, out, E);
}